// RBFPolicy_12455405158906
// MI455X (gfx1250) — compile-verified
//
#include <hip/hip_runtime.h>
#include <hip/hip_bf16.h>
#include <math.h>

// ---------------------------------------------------------------------------
// RBF GP policy evaluation on gfx1250 (MI455X).
//   v = K^{-1} Y solved with fixed-iteration CG (K SPD = RBF gram + 1e-4 I)
//   K*p applied matrix-free; 16x16 gram tiles via V_WMMA_F32_16X16X4_F32.
//   exp2-domain trick: A operand pre-scaled by log2(e), row term -log2(e)*h_i
//   seeded into the WMMA accumulator, so the MMA output feeds v_exp_f32
//   directly.  Dims stored permuted [0,1,4,5,2,3,6,7] so each lane's four
//   K-slots are one contiguous b128 load.
// ---------------------------------------------------------------------------

constexpr int   N      = 8192;
constexpr int   D      = 5;
constexpr int   DP     = 8;       // pad 5 -> 8 so two K=4 WMMA cover the dot
constexpr int   ITERS  = 512;     // ~sqrt(cond) for sigma^2 = 1e-4 noise floor
constexpr int   CSPLIT = 8;       // column chunks per 16-row strip
constexpr float SIGMA2 = 0.01f * 0.01f;
constexpr float LOG2E  = 1.44269504088896340736f;

typedef __attribute__((ext_vector_type(2))) float v2f;
typedef __attribute__((ext_vector_type(4))) float v4f;
typedef __attribute__((ext_vector_type(8))) float v8f;

// Storage permutation: mem slot j holds logical contraction slot PERM[j].
// half 0 lanes consume mem[0..3] = logical {0,1,4,5}; half 1 -> mem[4..7] =
// logical {2,3,6,7}.  Contraction is invariant since A and B share the order.
__device__ __constant__ int PERM_[1]; // (unused placeholder to keep constexpr local)

// ---------------------------------------------------------------------------
// Stage 0: precompute permuted scaled coordinates XspA (x log2e) and XspB,
// h = 0.5*|Xs|^2, cross kernel kvec, CG state r=p=Y, v=0, q=exp(-h)*p.
// ---------------------------------------------------------------------------
__global__ void rbf_init_kernel(const float* __restrict__ x,
                                const float* __restrict__ X,
                                const float* __restrict__ log_l,
                                const float* __restrict__ Y,
                                float* __restrict__ XspA,
                                float* __restrict__ XspB,
                                float* __restrict__ hs,
                                float* __restrict__ kvec,
                                float* __restrict__ v,
                                float* __restrict__ r,
                                float* __restrict__ p,
                                float* __restrict__ q) {
  const int perm[DP] = {0, 1, 4, 5, 2, 3, 6, 7};
  int n = blockIdx.x * blockDim.x + threadIdx.x;
  if (n >= N) return;
  float invl[D], s[D];
#pragma unroll
  for (int d = 0; d < D; ++d) {
    invl[d] = __expf(-2.0f * log_l[d]);
    s[d]    = sqrtf(invl[d]);
  }
  float xs[DP];
  float sq = 0.0f, dq = 0.0f;
#pragma unroll
  for (int d = 0; d < DP; ++d) {
    xs[d] = (d < D) ? X[n * D + d] * s[d] : 0.0f;
    sq = fmaf(xs[d], xs[d], sq);
    if (d < D) {
      float dd = x[d] - X[n * D + d];
      dq = fmaf(dd * dd, invl[d], dq);
    }
  }
#pragma unroll
  for (int j = 0; j < DP; ++j) {
    float t = xs[perm[j]];
    XspB[n * DP + j] = t;
    XspA[n * DP + j] = t * LOG2E;
  }
  float h = 0.5f * sq;
  hs[n]   = h;
  kvec[n] = __expf(-0.5f * dq);
  float y = Y[n];
  r[n] = y;
  p[n] = y;
  q[n] = __expf(-h) * y;
  v[n] = 0.0f;
}

// ---------------------------------------------------------------------------
// Partial matvec: Apart[chunk][i] = sum_{j in chunk} exp(G_ij - h_i) * q_j
// with q_j = exp(-h_j) p_j precomputed.  One wave = 16 rows x (N/CSPLIT) cols.
// WMMA directly yields log2e*(G_ij - h_i); v_exp_f32 (exp2) finishes it.
// ---------------------------------------------------------------------------
__global__ void __launch_bounds__(256)
rbf_matvec_kernel(const float* __restrict__ XspA,
                  const float* __restrict__ XspB,
                  const float* __restrict__ hs,
                  const float* __restrict__ q,
                  float* __restrict__ Apart) {
  const int lane  = threadIdx.x & 31;
  const int gw    = blockIdx.x * (blockDim.x >> 5) + (threadIdx.x >> 5);
  const int chunk = gw & (CSPLIT - 1);
  const int strip = gw >> 3;            // log2(CSPLIT) == 3
  const int rowBase  = strip * 16;
  const int colStart = chunk * (N / CSPLIT);
  const int colEnd   = colStart + (N / CSPLIT);
  const int half  = lane >> 4;          // 0: lanes 0-15, 1: lanes 16-31
  const int laneN = lane & 15;

  // A operand (16 rows x 8 dims, log2e-scaled): M = lane%16, K-slots for this
  // half are one contiguous b128 at offset 4*half thanks to the permutation.
  v4f av = *(const v4f*)&XspA[(rowBase + laneN) * DP + 4 * half];
  v2f a0 = {av.x, av.y};
  v2f a1 = {av.z, av.w};

  // Accumulator seed: C[m] = -log2e * h_row(m)  (constant over lanes).
  v8f c0;
#pragma unroll
  for (int m = 0; m < 8; ++m) c0[m] = -LOG2E * hs[rowBase + m + 8 * half];

  float acc[8];
#pragma unroll
  for (int m = 0; m < 8; ++m) acc[m] = 0.0f;

  for (int colBase = colStart; colBase < colEnd; colBase += 16) {
    v4f bv = *(const v4f*)&XspB[(colBase + laneN) * DP + 4 * half];
    v2f b0 = {bv.x, bv.y};
    v2f b1 = {bv.z, bv.w};

    v8f g = __builtin_amdgcn_wmma_f32_16x16x4_f32(false, a0, false, b0,
                                                  (short)0, c0, false, false);
    g = __builtin_amdgcn_wmma_f32_16x16x4_f32(false, a1, false, b1,
                                              (short)0, g, false, false);

    float qc = q[colBase + laneN];    // exp(-h_j) p_j, constant over m
#pragma unroll
    for (int m = 0; m < 8; ++m)
      acc[m] = fmaf(__builtin_amdgcn_exp2f(g[m]), qc, acc[m]);
  }

  // Sum over the 16 columns held by this half-wave.
#pragma unroll
  for (int m = 0; m < 8; ++m) {
    float a = acc[m];
    a += __shfl_xor(a, 1, 32);
    a += __shfl_xor(a, 2, 32);
    a += __shfl_xor(a, 4, 32);
    a += __shfl_xor(a, 8, 32);
    acc[m] = a;
  }
  if (laneN == 0) {
    int base = chunk * N + rowBase + 8 * half;
#pragma unroll
    for (int m = 0; m < 8; ++m) Apart[base + m] = acc[m];
  }
}

// Ap[i] = sum_c Apart[c][i] + sigma^2 p[i]   (deterministic chunk reduce)
__global__ void rbf_reduce_kernel(const float* __restrict__ Apart,
                                  const float* __restrict__ p,
                                  float* __restrict__ Ap) {
  int n = blockIdx.x * blockDim.x + threadIdx.x;
  if (n >= N) return;
  float s = 0.0f;
#pragma unroll
  for (int c = 0; c < CSPLIT; ++c) s += Apart[c * N + n];
  Ap[n] = fmaf(SIGMA2, p[n], s);
}

// ---------------------------------------------------------------------------
// Single-block dot product -> scalar in workspace (deterministic tree reduce).
// ---------------------------------------------------------------------------
__global__ void rbf_dot_kernel(const float* __restrict__ a,
                               const float* __restrict__ b,
                               float* __restrict__ out) {
  __shared__ float sm[256];
  float s = 0.0f;
  for (int i = threadIdx.x; i < N; i += 256) s = fmaf(a[i], b[i], s);
  sm[threadIdx.x] = s;
  __syncthreads();
  for (int w = 128; w > 0; w >>= 1) {
    if (threadIdx.x < w) sm[threadIdx.x] += sm[threadIdx.x + w];
    __syncthreads();
  }
  if (threadIdx.x == 0) *out = sm[0];
}

// alpha = rr/pAp ; v += alpha p ; r -= alpha Ap
__global__ void rbf_update1_kernel(const float* __restrict__ rr,
                                   const float* __restrict__ pAp,
                                   const float* __restrict__ p,
                                   const float* __restrict__ Ap,
                                   float* __restrict__ v,
                                   float* __restrict__ r) {
  int n = blockIdx.x * blockDim.x + threadIdx.x;
  if (n >= N) return;
  float alpha = *rr / *pAp;
  v[n] = fmaf(alpha, p[n], v[n]);
  r[n] = fmaf(-alpha, Ap[n], r[n]);
}

// beta = rr_new/rr_old ; p = r + beta p ; q = exp(-h) p
__global__ void rbf_update2_kernel(const float* __restrict__ rr_new,
                                   const float* __restrict__ rr_old,
                                   const float* __restrict__ r,
                                   const float* __restrict__ hs,
                                   float* __restrict__ p,
                                   float* __restrict__ q) {
  int n = blockIdx.x * blockDim.x + threadIdx.x;
  if (n >= N) return;
  float beta = *rr_new / *rr_old;
  float pn = fmaf(beta, p[n], r[n]);
  p[n] = pn;
  q[n] = __expf(-hs[n]) * pn;
}

// out = U_MAX * (9 sin(t) + sin(3t)) / 8,  t = kvec . v
__global__ void rbf_final_kernel(const float* __restrict__ kvec,
                                 const float* __restrict__ v,
                                 float* __restrict__ out) {
  __shared__ float sm[256];
  float s = 0.0f;
  for (int i = threadIdx.x; i < N; i += 256) s = fmaf(kvec[i], v[i], s);
  sm[threadIdx.x] = s;
  __syncthreads();
  for (int w = 128; w > 0; w >>= 1) {
    if (threadIdx.x < w) sm[threadIdx.x] += sm[threadIdx.x + w];
    __syncthreads();
  }
  if (threadIdx.x == 0) {
    float t = sm[0];
    out[0] = 10.0f * (9.0f * __sinf(t) + __sinf(3.0f * t)) * 0.125f;
  }
}

// ---------------------------------------------------------------------------
extern "C" void kernel_launch(void* const* d_in, const int* in_sizes, int n_in,
                              void* d_out, int out_size, void* d_ws, size_t ws_size,
                              hipStream_t stream) {
  (void)in_sizes; (void)n_in; (void)out_size; (void)ws_size;
  const float* x     = (const float*)d_in[0];   // (5,)
  const float* X     = (const float*)d_in[1];   // (8192, 5)
  const float* log_l = (const float*)d_in[2];   // (5,)
  const float* Y     = (const float*)d_in[3];   // (8192,)
  float* out = (float*)d_out;

  float* ws    = (float*)d_ws;
  float* XspA  = ws;               // N*DP floats (log2e-scaled, permuted)
  float* XspB  = XspA + N * DP;    // N*DP floats (permuted)
  float* hs    = XspB + N * DP;    // N
  float* kvec  = hs + N;           // N
  float* v     = kvec + N;         // N
  float* r     = v + N;            // N
  float* p     = r + N;            // N
  float* q     = p + N;            // N
  float* Ap    = q + N;            // N
  float* Apart = Ap + N;           // CSPLIT*N
  float* scal  = Apart + CSPLIT * N;  // [0],[1] = rr slots, [2] = pAp

  const int TPB = 256;
  // 512 row-strips x CSPLIT column chunks, 8 waves per 256-thread block.
  const int MV_BLOCKS = (N / 16) * CSPLIT / 8;

  rbf_init_kernel<<<N / TPB, TPB, 0, stream>>>(x, X, log_l, Y,
                                               XspA, XspB, hs, kvec,
                                               v, r, p, q);
  rbf_dot_kernel<<<1, TPB, 0, stream>>>(r, r, &scal[0]);

  int cur = 0;
  for (int it = 0; it < ITERS; ++it) {
    rbf_matvec_kernel<<<MV_BLOCKS, TPB, 0, stream>>>(XspA, XspB, hs, q, Apart);
    rbf_reduce_kernel<<<N / TPB, TPB, 0, stream>>>(Apart, p, Ap);
    rbf_dot_kernel<<<1, TPB, 0, stream>>>(p, Ap, &scal[2]);
    rbf_update1_kernel<<<N / TPB, TPB, 0, stream>>>(&scal[cur], &scal[2],
                                                    p, Ap, v, r);
    int nxt = 1 - cur;
    rbf_dot_kernel<<<1, TPB, 0, stream>>>(r, r, &scal[nxt]);
    rbf_update2_kernel<<<N / TPB, TPB, 0, stream>>>(&scal[nxt], &scal[cur],
                                                    r, hs, p, q);
    cur = nxt;
  }

  rbf_final_kernel<<<1, TPB, 0, stream>>>(kvec, v, out);
}